// FasterRCNNFeatureExtractor_1434519076998
// MI455X (gfx1250) — compile-verified
//
#include <hip/hip_runtime.h>

// Problem constants (from setup_inputs in the reference)
#define B_    16
#define C_    2048
#define H_    50
#define W_    50
#define HW_   2500
#define NDET  100
#define KOBJ  36      // num_objects
#define KPAD  48      // padded to 3 WMMA tiles of 16
#define CT_   (C_/16) // 128 channel tiles

typedef __attribute__((ext_vector_type(2))) float v2f;
typedef __attribute__((ext_vector_type(8))) float v8f;

// ---------------------------------------------------------------------------
// Kernel 1: box selection (reproduces both jnp.argsort orderings), region
// computation, spatial-feature output, and per-slot scale (= valid/area).
// ---------------------------------------------------------------------------
__global__ __launch_bounds__(128)
void frcnn_select_kernel(const float* __restrict__ boxes,
                         const float* __restrict__ scores,
                         int4*  __restrict__ ws_region,
                         float* __restrict__ ws_scale,
                         float* __restrict__ out_spatial) {
  const int b = blockIdx.x;
  const int t = threadIdx.x;

  __shared__ float s_sc[NDET];
  __shared__ int   s_v[NDET];
  __shared__ int   s_nv;

  if (t == 0) s_nv = 0;
  __syncthreads();
  if (t < NDET) {
    float sc = scores[b * NDET + t];
    int v = (sc > 0.5f) ? 1 : 0;
    s_sc[t] = sc;
    s_v[t]  = v;
    if (v) atomicAdd(&s_nv, 1);
  }
  __syncthreads();
  const int nv = s_nv;

  if (t < NDET) {
    const int   v  = s_v[t];
    const float sc = s_sc[t];
    int r = 0;
    if (v && (nv > KOBJ)) {
      // top-k by score (stable: index tiebreak), valid entries only
      for (int j = 0; j < NDET; ++j)
        if (s_v[j] && (s_sc[j] > sc || (s_sc[j] == sc && j < t))) ++r;
    } else if (v) {
      // original order among valid entries
      for (int j = 0; j < t; ++j) r += s_v[j];
    } else {
      // invalid entries trail in original order (both orderings)
      r = nv;
      for (int j = 0; j < t; ++j) r += 1 - s_v[j];
    }

    if (r < KOBJ) {
      const float4 bx = reinterpret_cast<const float4*>(boxes)[b * NDET + t];
      // match reference: floor(coord * W / 800) with the same fp sequence
      int fx1 = min(max((int)floorf((bx.x * (float)W_) / 800.0f), 0), W_ - 1);
      int fy1 = min(max((int)floorf((bx.y * (float)H_) / 800.0f), 0), H_ - 1);
      int fx2 = max(fx1 + 1, min((int)floorf((bx.z * (float)W_) / 800.0f), W_));
      int fy2 = max(fy1 + 1, min((int)floorf((bx.w * (float)H_) / 800.0f), H_));

      int4  reg;
      float scale;
      if (v) {
        reg   = make_int4(fx1, fy1, fx2, fy2);
        scale = 1.0f / (float)((fx2 - fx1) * (fy2 - fy1));
      } else {
        reg   = make_int4(0, 0, 0, 0);   // empty mask -> pooled = 0
        scale = 0.0f;                    // zeroes features of invalid slots
      }
      ws_region[b * KPAD + r] = reg;
      ws_scale[b * KPAD + r]  = scale;

      float4 ob = v ? bx : make_float4(0.f, 0.f, 0.f, 0.f);
      reinterpret_cast<float4*>(out_spatial)[b * KOBJ + r] = ob;
    }
  } else if (t < NDET + (KPAD - KOBJ)) {
    // padding slots 36..47: empty region, zero scale
    const int slot = KOBJ + (t - NDET);
    ws_region[b * KPAD + slot] = make_int4(0, 0, 0, 0);
    ws_scale[b * KPAD + slot]  = 0.0f;
  }
}

// ---------------------------------------------------------------------------
// Kernel 2: pooled features as feat[16,HW] x mask[HW,16] tiles with
// V_WMMA_F32_16X16X4_F32. One block per (batch, 16-channel slab); the slab
// (160 KB, contiguous in memory) is staged once in LDS (2 blocks/WGP);
// 3 waves cover the 48 (padded) box slots. The rectangle mask is generated
// from precomputed 50-bit column/row bitmasks: ~6 VALU per WMMA.
// ---------------------------------------------------------------------------
__global__ __launch_bounds__(96)
void frcnn_pool_kernel(const float* __restrict__ feat,
                       const int4*  __restrict__ ws_region,
                       const float* __restrict__ ws_scale,
                       float* __restrict__ out_feat,
                       int nblocks) {
  extern __shared__ float tile[];                 // 16 * 2500 floats = 160 KB
  const int b  = blockIdx.x >> 7;                 // / 128
  const int c0 = (blockIdx.x & 127) << 4;         // channel-tile base
  const int tid = threadIdx.x;

  // 16 consecutive C planes are contiguous: one linear 160 KB copy to LDS.
  const float4* src = reinterpret_cast<const float4*>(
      feat + (size_t)(b * C_ + c0) * HW_);
  float4* dst = reinterpret_cast<float4*>(tile);
  for (int i = tid; i < (16 * HW_) / 4; i += 96) dst[i] = src[i];

  // Warm L2 for the next block's slab (contiguous, stream order). Speculative
  // prefetch is counter-free and silently dropped on bad translation.
  if (blockIdx.x + 1 < (unsigned)nblocks) {
    const char* nxt = reinterpret_cast<const char*>(src) + 16 * HW_ * 4;
    for (int i = tid * 128; i < 16 * HW_ * 4; i += 96 * 128)
      __builtin_prefetch(nxt + i, 0, 1);
  }
  __syncthreads();

  const int lane = tid & 31;
  const int kt   = tid >> 5;            // wave id = K tile (0..2)
  const int n    = lane & 15;           // N column (and A's M row) per lane
  const int off  = (lane >> 4) << 1;    // 0 or 2 (reduction sub-offset)

  // per-lane box region for column k = kt*16 + n -> bitmask registers
  const int4 reg = ws_region[b * KPAD + kt * 16 + n];
  const unsigned long long cb = (1ull << reg.z) - (1ull << reg.x); // col bits
  unsigned long long rb       = (1ull << reg.w) - (1ull << reg.y); // row bits
  unsigned long long cbrow    = (rb & 1ull) ? cb : 0ull;           // row 0

  const float* arow = tile + n * HW_ + off;       // 8B-aligned (2500, off even)
  int x = off;                                    // (hw+off) mod 50; even
  v8f acc = {0.f, 0.f, 0.f, 0.f, 0.f, 0.f, 0.f, 0.f};

#pragma unroll 5
  for (int hw = 0; hw < HW_; hw += 4) {
    // A fragment: lane holds feat[c0+n][hw+off .. hw+off+1]
    v2f a = *reinterpret_cast<const v2f*>(arow + hw);
    // B fragment: two mask bits from the current row's column bitmask.
    // x is even, so the (x, x+1) pair never crosses a row of width 50.
    const unsigned int bits = (unsigned int)(cbrow >> x) & 3u;
    v2f bb;
    bb.x = (float)(bits & 1u);
    bb.y = (float)(bits >> 1);
    acc = __builtin_amdgcn_wmma_f32_16x16x4_f32(
        /*neg_a=*/false, a, /*neg_b=*/false, bb,
        /*c_mod=*/(short)0, acc, /*reuse_a=*/false, /*reuse_b=*/false);
    x += 4;
    if (x >= W_) {                      // row transition (every ~13 iters)
      x -= W_;
      rb >>= 1;
      cbrow = (rb & 1ull) ? cb : 0ull;
    }
  }

  // C/D layout: lanes 0-15 hold M=0..7 (VGPR v), lanes 16-31 hold M=8..15.
  const int k = kt * 16 + n;
  if (k < KOBJ) {
    const float s = ws_scale[b * KPAD + k];       // valid/area or 0
    float* o = out_feat + (size_t)(b * KOBJ + k) * C_ + c0 + ((lane >> 4) << 3);
    float4 lo = make_float4(acc[0] * s, acc[1] * s, acc[2] * s, acc[3] * s);
    float4 hi = make_float4(acc[4] * s, acc[5] * s, acc[6] * s, acc[7] * s);
    *reinterpret_cast<float4*>(o)     = lo;
    *reinterpret_cast<float4*>(o + 4) = hi;
  }
}

// ---------------------------------------------------------------------------
extern "C" void kernel_launch(void* const* d_in, const int* in_sizes, int n_in,
                              void* d_out, int out_size, void* d_ws, size_t ws_size,
                              hipStream_t stream) {
  const float* feat   = (const float*)d_in[0];   // [16,2048,50,50] f32
  const float* boxes  = (const float*)d_in[1];   // [16,100,4] f32
  const float* scores = (const float*)d_in[2];   // [16,100] f32
  // d_in[3] = num_objects (== 36, from setup_inputs; baked into tiling)

  float* out_feat = (float*)d_out;                              // [16,36,2048]
  float* out_sp   = (float*)d_out + (size_t)B_ * KOBJ * C_;     // [16,36,4]

  int4*  ws_region = (int4*)d_ws;                               // 16*48 int4
  float* ws_scale  = (float*)((char*)d_ws + (size_t)B_ * KPAD * sizeof(int4));

  frcnn_select_kernel<<<B_, 128, 0, stream>>>(boxes, scores, ws_region,
                                              ws_scale, out_sp);

  const int nblocks = B_ * CT_;
  frcnn_pool_kernel<<<nblocks, 96, 16 * HW_ * sizeof(float), stream>>>(
      feat, ws_region, ws_scale, out_feat, nblocks);
}